// WeightedGCN_64089501991220
// MI455X (gfx1250) — compile-verified
//
#include <hip/hip_runtime.h>

// WeightedGCN (2-layer GCNConv, D=64 everywhere) for MI455X / gfx1250.
// GEMMs use V_WMMA_F32_16X16X4_F32 (exact fp32 matrix cores, wave32 layout);
// edge aggregation uses L2-resident f32 global atomics (all feature matrices
// fit in the 192MB L2, so the scatter never leaves the cache).

typedef __attribute__((ext_vector_type(2))) float v2f;
typedef __attribute__((ext_vector_type(8))) float v8f;

// ---- degree accumulation: deg[dst[e]] += w[e] -------------------------------
__global__ void gcn_deg_accum(const int* __restrict__ dst, const float* __restrict__ w,
                              float* __restrict__ deg, int E) {
  int e = blockIdx.x * blockDim.x + threadIdx.x;
  if (e < E) atomicAdd(&deg[dst[e]], w[e]);
}

// ---- dis[n] = rsqrt(deg[n] + 1)  (self-loop weight 1; deg>=1 always) --------
__global__ void gcn_dis(float* __restrict__ deg_dis, int N) {
  int n = blockIdx.x * blockDim.x + threadIdx.x;
  if (n < N) deg_dis[n] = rsqrtf(deg_dis[n] + 1.0f);
}

// ---- H = X @ W^T  (X:[N,64] row-major, W:[64,64] stored [out,in]) -----------
// Block = 256 threads = 8 waves. Wave w: row-tile = w&1 (16 nodes),
// col-tile = w>>1 (16 outputs). K=64 via 16 chained 16x16x4 f32 WMMAs.
// Requires N % 32 == 0 (holds: N = 100000). No divergence -> EXEC all ones.
__global__ void gcn_gemm_nt64(const float* __restrict__ X, const float* __restrict__ W,
                              float* __restrict__ H) {
  const int wave  = threadIdx.x >> 5;
  const int lane  = threadIdx.x & 31;
  const int mbase = (blockIdx.x * 2 + (wave & 1)) * 16;   // node row base
  const int obase = (wave >> 1) * 16;                     // output col base
  const int lm    = lane & 15;
  const int lk    = (lane >> 4) << 1;                     // 0 or 2 within K-quad

  const float* __restrict__ xrow = X + (size_t)(mbase + lm) * 64; // A[M=lm][k]
  const float* __restrict__ wrow = W + (size_t)(obase + lm) * 64; // B[k][N=lm] = W[o][k]

  v8f acc = {0.f, 0.f, 0.f, 0.f, 0.f, 0.f, 0.f, 0.f};
#pragma unroll
  for (int kt = 0; kt < 16; ++kt) {
    const int k = kt * 4 + lk;                 // even -> 8B aligned
    v2f a = *(const v2f*)(xrow + k);           // A frag: {A[m][k], A[m][k+1]}
    v2f b = *(const v2f*)(wrow + k);           // B frag: {B[k][n], B[k+1][n]}
    acc = __builtin_amdgcn_wmma_f32_16x16x4_f32(
        false, a, false, b, (short)0, acc, false, false);
  }

  // C/D layout: VGPR r -> row (r + 8*lane[4]), col = lane&15.
  float* __restrict__ hout = H + (size_t)(mbase + ((lane >> 4) << 3)) * 64 + obase + lm;
#pragma unroll
  for (int r = 0; r < 8; ++r) hout[(size_t)r * 64] = acc[r];
}

// ---- edge scatter: G[dst] += H[src] * (dis[src]*w*dis[dst]) -----------------
// 64 consecutive threads per edge -> 256B coalesced gather + atomic burst.
__global__ void gcn_scatter(const float* __restrict__ H, const int* __restrict__ src,
                            const int* __restrict__ dst, const float* __restrict__ w,
                            const float* __restrict__ dis, float* __restrict__ G, int E) {
  int gid = blockIdx.x * blockDim.x + threadIdx.x;  // max E*64 = 80M < 2^31
  int e = gid >> 6;
  int f = gid & 63;
  if (e < E) {
    int s = src[e], d = dst[e];
    float coef = dis[s] * w[e] * dis[d];
    atomicAdd(&G[(size_t)d * 64 + f], H[(size_t)s * 64 + f] * coef);
  }
}

// ---- layer-1 epilogue: G = relu(G + H*dis^2 + b) ---------------------------
__global__ void gcn_post_relu(float* __restrict__ G, const float* __restrict__ H,
                              const float* __restrict__ dis, const float* __restrict__ b,
                              int total) {
  int i = blockIdx.x * blockDim.x + threadIdx.x;
  if (i < total) {
    int n = i >> 6, f = i & 63;
    float di = dis[n];
    float v  = G[i] + H[i] * di * di + b[f];
    G[i] = v > 0.f ? v : 0.f;
  }
}

// ---- layer-2 epilogue: O += H*dis^2 + b ------------------------------------
__global__ void gcn_post_final(float* __restrict__ O, const float* __restrict__ H,
                               const float* __restrict__ dis, const float* __restrict__ b,
                               int total) {
  int i = blockIdx.x * blockDim.x + threadIdx.x;
  if (i < total) {
    int n = i >> 6, f = i & 63;
    float di = dis[n];
    O[i] = O[i] + H[i] * di * di + b[f];
  }
}

extern "C" void kernel_launch(void* const* d_in, const int* in_sizes, int n_in,
                              void* d_out, int out_size, void* d_ws, size_t ws_size,
                              hipStream_t stream) {
  const float* x  = (const float*)d_in[0];
  const int*   ei = (const int*)  d_in[1];
  const float* w  = (const float*)d_in[2];
  const float* W1 = (const float*)d_in[3];
  const float* b1 = (const float*)d_in[4];
  const float* W2 = (const float*)d_in[5];
  const float* b2 = (const float*)d_in[6];

  const int N = in_sizes[0] / 64;   // 100000 (multiple of 32)
  const int E = in_sizes[2];        // 1250000
  const int* src = ei;              // edge_index[0]
  const int* dst = ei + E;          // edge_index[1]

  // workspace: dis[N] | H[N*64] | G[N*64]  (~51.6 MB)
  float* dis = (float*)d_ws;
  float* H   = dis + N;
  float* G   = H + (size_t)N * 64;
  float* out = (float*)d_out;

  const int    total      = N * 64;
  const size_t feat_bytes = (size_t)total * sizeof(float);
  const int    sgrid      = (int)(((long long)E * 64 + 255) / 256);

  // degree + normalization (shared by both layers)
  hipMemsetAsync(dis, 0, (size_t)N * sizeof(float), stream);
  gcn_deg_accum<<<(E + 255) / 256, 256, 0, stream>>>(dst, w, dis, E);
  gcn_dis<<<(N + 255) / 256, 256, 0, stream>>>(dis, N);

  // layer 1
  gcn_gemm_nt64<<<N / 32, 256, 0, stream>>>(x, W1, H);
  hipMemsetAsync(G, 0, feat_bytes, stream);
  gcn_scatter<<<sgrid, 256, 0, stream>>>(H, src, dst, w, dis, G, E);
  gcn_post_relu<<<(total + 255) / 256, 256, 0, stream>>>(G, H, dis, b1, total);

  // layer 2 (aggregate directly into d_out)
  gcn_gemm_nt64<<<N / 32, 256, 0, stream>>>(G, W2, H);
  hipMemsetAsync(out, 0, feat_bytes, stream);
  gcn_scatter<<<sgrid, 256, 0, stream>>>(H, src, dst, w, dis, out, E);
  gcn_post_final<<<(total + 255) / 256, 256, 0, stream>>>(out, H, dis, b2, total);
}